// IDWT_88347477279322
// MI455X (gfx1250) — compile-verified
//
#include <hip/hip_runtime.h>

// Inverse Haar DWT, specialized for x:(4,256,256,256) f32 -> out:(4,512,512,64) f32.
// Memory-bound streaming kernel: 536 MB total traffic, ~23 us floor at 23.3 TB/s.
// Strategy: b128 loads/stores with non-temporal cache hints (gfx1250 TH_NT path),
// 16 lanes per input pixel so every VMEM op is a coalesced 256B segment per half-wave.

typedef float v4f __attribute__((ext_vector_type(4)));

#define B_DIM 4
#define H_DIM 256
#define W_DIM 256
#define C_DIM 256   // input channels; output channels = 64

__global__ __launch_bounds__(256) void idwt_haar_stream(
    const float* __restrict__ x, float* __restrict__ out)
{
    // 16 threads per input pixel: 64 output channels as 16 x float4.
    const int tid = blockIdx.x * 256 + threadIdx.x;
    const int c4  = tid & 15;    // float4 index within 64-channel group
    const int pix = tid >> 4;    // (b*256 + h)*256 + w

    const int w = pix & 255;
    const int h = (pix >> 8) & 255;
    const int b = pix >> 16;

    // Input pixel base is 1024B-aligned; 4 contiguous 64-channel groups.
    const v4f* __restrict__ pin = (const v4f*)(x + (size_t)pix * (size_t)C_DIM);
    const v4f a = __builtin_nontemporal_load(pin + 0 * 16 + c4);  // x1 group
    const v4f e = __builtin_nontemporal_load(pin + 1 * 16 + c4);  // x2 group
    const v4f c = __builtin_nontemporal_load(pin + 2 * 16 + c4);  // x3 group
    const v4f d = __builtin_nontemporal_load(pin + 3 * 16 + c4);  // x4 group

    // Butterfly (0.5 scale folded to the end):
    //   h01 = 0.5*(a - e - c + d)   -> out[2h,   2w]
    //   h02 = 0.5*(a + e - c - d)   -> out[2h+1, 2w]
    //   h03 = 0.5*(a - e + c - d)   -> out[2h,   2w+1]
    //   h04 = 0.5*(a + e + c + d)   -> out[2h+1, 2w+1]
    const v4f s1 = a + e;
    const v4f s2 = a - e;
    const v4f s3 = c + d;
    const v4f s4 = c - d;

    const v4f h01 = 0.5f * (s2 - s4);
    const v4f h02 = 0.5f * (s1 - s3);
    const v4f h03 = 0.5f * (s2 + s4);
    const v4f h04 = 0.5f * (s1 + s3);

    // Output geometry: rows of 512*64 = 32768 floats = 8192 float4.
    // Row r0 = b*512 + 2h; columns 2w (then 2w+1 at +16 float4).
    const int    r0   = b * 512 + 2 * h;
    v4f* __restrict__ o0 = (v4f*)out + (size_t)r0 * 8192 + (size_t)w * 32 + c4;
    v4f* __restrict__ o1 = o0 + 8192;  // row 2h+1

    __builtin_nontemporal_store(h01, o0);        // [2h,   2w]
    __builtin_nontemporal_store(h03, o0 + 16);   // [2h,   2w+1]
    __builtin_nontemporal_store(h02, o1);        // [2h+1, 2w]
    __builtin_nontemporal_store(h04, o1 + 16);   // [2h+1, 2w+1]
}

extern "C" void kernel_launch(void* const* d_in, const int* in_sizes, int n_in,
                              void* d_out, int out_size, void* d_ws, size_t ws_size,
                              hipStream_t stream)
{
    (void)in_sizes; (void)n_in; (void)out_size; (void)d_ws; (void)ws_size;
    const float* x   = (const float*)d_in[0];
    float*       out = (float*)d_out;

    // 4*256*256 pixels * 16 threads each = 4,194,304 threads.
    const int total_threads = B_DIM * H_DIM * W_DIM * 16;
    const int block = 256;                       // 8 wave32 waves per workgroup
    const int grid  = total_threads / block;     // 16384 blocks

    idwt_haar_stream<<<grid, block, 0, stream>>>(x, out);
}